// CachedMultiHeadAttention_84997402788179
// MI455X (gfx1250) — compile-verified
//
#include <hip/hip_runtime.h>
#include <hip/hip_bf16.h>
#include <math.h>

#define D_MODEL   4096
#define NUM_HEADS 32
#define HEAD_DIM  128
#define MAX_SEQ   32768
#define TQ        16
#define NSPLIT    32
#define NW        2
#define KS        132            // padded LDS row stride (floats): 16B-aligned rows, bank-spread
#define INV_SCALE 0.08838834764831845f  // 1/sqrt(128)
#define NEG_BIG   -3.0e38f

typedef float v8f __attribute__((ext_vector_type(8)));
typedef float v2f __attribute__((ext_vector_type(2)));
typedef int   v4i __attribute__((ext_vector_type(4)));

typedef __attribute__((address_space(3))) float lds_f32;
typedef __attribute__((address_space(1))) v4i*  gv4i_p;   // global v4i*
typedef __attribute__((address_space(3))) v4i*  lv4i_p;   // LDS v4i*

static __device__ __forceinline__ v8f wmma4(v2f a, v2f b, v8f c) {
  // D(16x16,f32) = A(16x4,f32) * B(4x16,f32) + C
  return __builtin_amdgcn_wmma_f32_16x16x4_f32(
      false, a, false, b, (short)0, c, false, false);
}

// ---- CDNA5 async global->LDS copy (ASYNCcnt), builtin with asm fallback ----
#if defined(__has_builtin) && __has_builtin(__builtin_amdgcn_global_load_async_to_lds_b128)
static __device__ __forceinline__ void async_copy16(const float* g, float* l) {
  __builtin_amdgcn_global_load_async_to_lds_b128((gv4i_p)g, (lv4i_p)l, 0, 0);
}
#else
static __device__ __forceinline__ void async_copy16(const float* g, float* l) {
  unsigned lds_off = (unsigned)(unsigned long long)(lds_f32*)l;
  unsigned long long ga = (unsigned long long)g;
  asm volatile("global_load_async_to_lds_b128 %0, %1, off"
               :: "v"(lds_off), "v"(ga) : "memory");
}
#endif

#if defined(__has_builtin) && __has_builtin(__builtin_amdgcn_s_wait_asynccnt)
#define WAIT_ASYNC(n) __builtin_amdgcn_s_wait_asynccnt(n)
#else
#define WAIT_ASYNC(n) asm volatile("s_wait_asynccnt %0" :: "i"(n) : "memory")
#endif

// ---------------------------------------------------------------------------
// Kernel 1: QKV projection.  One wave computes one 16x16 tile of X(16x4096)*W.
// ---------------------------------------------------------------------------
__global__ __launch_bounds__(128) void qkv_gemm_kernel(
    const float* __restrict__ x,
    const float* __restrict__ Wq, const float* __restrict__ bq,
    const float* __restrict__ Wk, const float* __restrict__ bk,
    const float* __restrict__ Wv, const float* __restrict__ bv,
    float* __restrict__ q_ws, float* __restrict__ kn_ws, float* __restrict__ vn_ws) {
  const int w    = threadIdx.x >> 5;
  const int lane = threadIdx.x & 31;
  const int tile = blockIdx.x * 4 + w;        // 0..767
  const int sel  = tile >> 8;                 // 0=Q,1=K,2=V
  const int cb   = tile & 255;

  const float* W    = (sel == 0) ? Wq : (sel == 1) ? Wk : Wv;
  const float* bias = (sel == 0) ? bq : (sel == 1) ? bk : bv;

  const int mn  = lane & 15;
  const int hl  = lane >> 4;
  const int col = cb * 16 + mn;

  v8f acc = {};
  for (int k0 = 0; k0 < D_MODEL; k0 += 32) {
    __builtin_prefetch(W + (size_t)(k0 + 64 + hl * 2) * D_MODEL + col, 0, 1);
#pragma unroll
    for (int k = k0; k < k0 + 32; k += 4) {
      v2f a = *(const v2f*)(x + mn * D_MODEL + k + hl * 2);
      v2f b;
      b[0] = W[(k + hl * 2)     * D_MODEL + col];
      b[1] = W[(k + hl * 2 + 1) * D_MODEL + col];
      acc = wmma4(a, b, acc);
    }
  }
  const float bb = bias[col];

  if (sel == 0) {
#pragma unroll
    for (int j = 0; j < 8; j++)
      q_ws[(j + hl * 8) * D_MODEL + col] = (acc[j] + bb) * INV_SCALE;
  } else {
    float* dst  = (sel == 1) ? kn_ws : vn_ws;
    const int h = col >> 7, d = col & 127;
#pragma unroll
    for (int j = 0; j < 8; j++)
      dst[(h * TQ + (j + hl * 8)) * HEAD_DIM + d] = acc[j] + bb;
  }
}

// ---------------------------------------------------------------------------
// Kernel 2: flash attention. 2 waves/workgroup, 32 splits/head.
// K/V tiles staged into LDS via async global->LDS (ASYNCcnt), WMMA f32 math.
// ---------------------------------------------------------------------------
__global__ __launch_bounds__(64) void attn_kernel(
    const float* __restrict__ q_ws,
    const float* __restrict__ kc, const float* __restrict__ vc,
    const float* __restrict__ kn, const float* __restrict__ vn,
    const int* __restrict__ posp,
    float* __restrict__ part_o, float* __restrict__ part_m, float* __restrict__ part_l) {
  __shared__ __align__(16) float sQ[TQ * KS];        // padded Q tile
  __shared__ __align__(16) float sK[NW][TQ * KS];    // per-wave K stage
  __shared__ __align__(16) float sV[NW][TQ * KS];    // per-wave V stage
  __shared__ float sP[NW][TQ * 18];                  // per-wave P transpose stage
  __shared__ float sO[TQ * HEAD_DIM];
  __shared__ float sLt[TQ];
  __shared__ float sM[NW][TQ];

  const int tid   = threadIdx.x;
  const int w     = tid >> 5;
  const int lane  = tid & 31;
  const int head  = blockIdx.x / NSPLIT;
  const int split = blockIdx.x % NSPLIT;
  const int pos   = *posp;
  const int L     = pos + TQ;
  const int nblk  = (L + 15) >> 4;

  for (int i = tid; i < TQ * HEAD_DIM; i += 64) {
    const int t = i >> 7, d = i & 127;
    sQ[t * KS + d] = q_ws[t * D_MODEL + head * HEAD_DIM + d];
    sO[i] = 0.0f;
  }
  if (tid < TQ) sLt[tid] = 0.0f;
  __syncthreads();

  const int mn = lane & 15;
  const int hl = lane >> 4;

  v2f qa[32];
#pragma unroll
  for (int t = 0; t < 32; t++)
    qa[t] = *(const v2f*)(sQ + mn * KS + 4 * t + hl * 2);

  const v8f vzero = {};
  v8f oacc[8];
#pragma unroll
  for (int i = 0; i < 8; i++) oacc[i] = vzero;
  float m_run[8], l_run[8];
#pragma unroll
  for (int j = 0; j < 8; j++) { m_run[j] = NEG_BIG; l_run[j] = 0.0f; }

  float* const sKw = sK[w];
  float* const sVw = sV[w];
  float* const sPw = sP[w];

  for (int b = split * NW + w; b < nblk; b += NSPLIT * NW) {
    const int kb = b * 16;

    // ---- stage K then V tiles (8KB each) into LDS asynchronously ----
    {
      const int srow  = lane >> 1;                 // 2 lanes per key row
      const int soff  = (lane & 1) * 4;            // float offset within row
      const int kid_s = (kb + srow < L) ? kb + srow : L - 1;
      const float* kg = (kid_s < pos)
          ? kc + (size_t)(head * MAX_SEQ + kid_s) * HEAD_DIM
          : kn + (size_t)(head * TQ + (kid_s - pos)) * HEAD_DIM;
      const float* vg = (kid_s < pos)
          ? vc + (size_t)(head * MAX_SEQ + kid_s) * HEAD_DIM
          : vn + (size_t)(head * TQ + (kid_s - pos)) * HEAD_DIM;
      float* kd = sKw + srow * KS + soff;
      float* vd = sVw + srow * KS + soff;
#pragma unroll
      for (int i = 0; i < 16; i++) async_copy16(kg + soff + i * 8, kd + i * 8);
#pragma unroll
      for (int i = 0; i < 16; i++) async_copy16(vg + soff + i * 8, vd + i * 8);
    }
    WAIT_ASYNC(16);   // K tile complete (async loads retire in order); V in flight

    // ---- S = (Q/sqrt(d)) * K^T (16x16) from LDS ----
    v8f s = vzero;
#pragma unroll
    for (int t = 0; t < 32; t++) {
      v2f kf = *(const v2f*)(sKw + mn * KS + 4 * t + hl * 2);
      s = wmma4(qa[t], kf, s);
    }
    const bool oob = (kb + mn >= L);

    // ---- online softmax (row r = j + hl*8 spans the 16 lanes of a half) ----
#pragma unroll
    for (int j = 0; j < 8; j++) {
      float sj = oob ? NEG_BIG : s[j];
      float mx = sj;
      mx = fmaxf(mx, __shfl_xor(mx, 1, 32));
      mx = fmaxf(mx, __shfl_xor(mx, 2, 32));
      mx = fmaxf(mx, __shfl_xor(mx, 4, 32));
      mx = fmaxf(mx, __shfl_xor(mx, 8, 32));
      const float mnew  = fmaxf(m_run[j], mx);
      const float alpha = (m_run[j] > -1.0e37f) ? __expf(m_run[j] - mnew) : 0.0f;
      const float pj    = (sj > -1.0e37f) ? __expf(sj - mnew) : 0.0f;
      float ps = pj;
      ps += __shfl_xor(ps, 1, 32);
      ps += __shfl_xor(ps, 2, 32);
      ps += __shfl_xor(ps, 4, 32);
      ps += __shfl_xor(ps, 8, 32);
      l_run[j] = l_run[j] * alpha + ps;
      m_run[j] = mnew;
#pragma unroll
      for (int nt = 0; nt < 8; nt++) oacc[nt][j] *= alpha;
      sPw[(j + hl * 8) * 18 + mn] = pj;    // per-wave LDS: in-order, no barrier
    }

    WAIT_ASYNC(0);    // V tile complete

    // ---- O += P(16x16) * V(16x128) from LDS ----
#pragma unroll
    for (int tt = 0; tt < 4; tt++) {
      v2f pa = *(const v2f*)(sPw + mn * 18 + 4 * tt + hl * 2);
      const int r0 = (4 * tt + hl * 2) * KS;
#pragma unroll
      for (int nt = 0; nt < 8; nt++) {
        v2f bf;
        bf[0] = sVw[r0 + nt * 16 + mn];
        bf[1] = sVw[r0 + KS + nt * 16 + mn];
        oacc[nt] = wmma4(pa, bf, oacc[nt]);
      }
    }
  }

  // ---- merge the 2 waves of this workgroup in LDS ----
  if (mn == 0) {
#pragma unroll
    for (int j = 0; j < 8; j++) sM[w][j + hl * 8] = m_run[j];
  }
  __syncthreads();
#pragma unroll
  for (int j = 0; j < 8; j++) {
    const int r = j + hl * 8;
    float M = fmaxf(sM[0][r], sM[1][r]);
    const float sc = (m_run[j] > -1.0e37f) ? __expf(m_run[j] - M) : 0.0f;
#pragma unroll
    for (int nt = 0; nt < 8; nt++)
      atomicAdd(&sO[r * HEAD_DIM + nt * 16 + mn], oacc[nt][j] * sc);
    if (mn == 0) atomicAdd(&sLt[r], l_run[j] * sc);
  }
  __syncthreads();

  const int base = head * NSPLIT + split;
  for (int i = tid; i < TQ * HEAD_DIM; i += 64)
    part_o[(size_t)base * TQ * HEAD_DIM + i] = sO[i];
  if (tid < TQ) {
    part_m[base * TQ + tid] = fmaxf(sM[0][tid], sM[1][tid]);
    part_l[base * TQ + tid] = sLt[tid];
  }
}

// ---------------------------------------------------------------------------
// Kernel 3: merge NSPLIT partials per head, normalize, write [t][4096] layout.
// ---------------------------------------------------------------------------
__global__ __launch_bounds__(256) void merge_kernel(
    const float* __restrict__ part_o, const float* __restrict__ part_m,
    const float* __restrict__ part_l, float* __restrict__ attn_ws) {
  const int h = blockIdx.x;
  for (int i = threadIdx.x; i < TQ * HEAD_DIM; i += 256) {
    const int r = i >> 7, d = i & 127;
    float M = NEG_BIG;
    for (int s = 0; s < NSPLIT; s++)
      M = fmaxf(M, part_m[(h * NSPLIT + s) * TQ + r]);
    float acc = 0.0f, lt = 0.0f;
    for (int s = 0; s < NSPLIT; s++) {
      const float ms = part_m[(h * NSPLIT + s) * TQ + r];
      const float sc = (ms > -1.0e37f) ? __expf(ms - M) : 0.0f;
      acc += sc * part_o[(size_t)(h * NSPLIT + s) * TQ * HEAD_DIM + i];
      lt  += sc * part_l[(h * NSPLIT + s) * TQ + r];
    }
    attn_ws[r * D_MODEL + h * HEAD_DIM + d] = acc / lt;
  }
}

// ---------------------------------------------------------------------------
// Kernel 4: output projection  out = attn(16x4096) * Wo + bo
// ---------------------------------------------------------------------------
__global__ __launch_bounds__(128) void out_gemm_kernel(
    const float* __restrict__ a_in, const float* __restrict__ Wo,
    const float* __restrict__ bo, float* __restrict__ out) {
  const int w    = threadIdx.x >> 5;
  const int lane = threadIdx.x & 31;
  const int cb   = blockIdx.x * 4 + w;   // 0..255
  const int mn   = lane & 15;
  const int hl   = lane >> 4;
  const int col  = cb * 16 + mn;

  v8f acc = {};
  for (int k0 = 0; k0 < D_MODEL; k0 += 32) {
    __builtin_prefetch(Wo + (size_t)(k0 + 64 + hl * 2) * D_MODEL + col, 0, 1);
#pragma unroll
    for (int k = k0; k < k0 + 32; k += 4) {
      v2f a = *(const v2f*)(a_in + mn * D_MODEL + k + hl * 2);
      v2f b;
      b[0] = Wo[(k + hl * 2)     * D_MODEL + col];
      b[1] = Wo[(k + hl * 2 + 1) * D_MODEL + col];
      acc = wmma4(a, b, acc);
    }
  }
  const float bb = bo[col];
#pragma unroll
  for (int j = 0; j < 8; j++)
    out[(j + hl * 8) * D_MODEL + col] = acc[j] + bb;
}

// ---------------------------------------------------------------------------
extern "C" void kernel_launch(void* const* d_in, const int* in_sizes, int n_in,
                              void* d_out, int out_size, void* d_ws, size_t ws_size,
                              hipStream_t stream) {
  const float* x    = (const float*)d_in[0];
  const float* kc   = (const float*)d_in[1];
  const float* vc   = (const float*)d_in[2];
  const float* Wq   = (const float*)d_in[3];
  const float* bq   = (const float*)d_in[4];
  const float* Wk   = (const float*)d_in[5];
  const float* bk   = (const float*)d_in[6];
  const float* Wv   = (const float*)d_in[7];
  const float* bv   = (const float*)d_in[8];
  const float* Wo   = (const float*)d_in[9];
  const float* bo   = (const float*)d_in[10];
  const int*   posp = (const int*)d_in[11];
  float*       out  = (float*)d_out;

  float* ws     = (float*)d_ws;
  float* q_ws   = ws;                        // 16*4096              = 65536
  float* kn_ws  = ws + 65536;                // 32*16*128            = 65536
  float* vn_ws  = ws + 131072;               // 65536
  float* part_o = ws + 196608;               // 32*32*16*128         = 2097152
  float* part_m = ws + 2293760;              // 32*32*16             = 16384
  float* part_l = ws + 2310144;              // 16384
  float* attn   = ws + 2326528;              // 65536   (total ~9.6 MB)

  qkv_gemm_kernel<<<192, 128, 0, stream>>>(x, Wq, bq, Wk, bk, Wv, bv,
                                           q_ws, kn_ws, vn_ws);
  attn_kernel<<<NUM_HEADS * NSPLIT, 64, 0, stream>>>(q_ws, kc, vc, kn_ws, vn_ws,
                                                     posp, part_o, part_m, part_l);
  merge_kernel<<<NUM_HEADS, 256, 0, stream>>>(part_o, part_m, part_l, attn);
  out_gemm_kernel<<<64, 128, 0, stream>>>(attn, Wo, bo, out);
}